// SpinorBlock_25142738551441
// MI455X (gfx1250) — compile-verified
//
#include <hip/hip_runtime.h>

typedef __attribute__((ext_vector_type(2))) float v2f;
typedef __attribute__((ext_vector_type(8))) float v8f;

#define BATCH 2
#define SEQ   4096
#define DIM   4096
#define BSZ   8      // block size
#define NBLK  512    // number of diagonal blocks
#define ROWS  (BATCH * SEQ)          // 8192
#define TILES_PER_WAVE 4             // 4 x 16 = 64 rows per wave
#define NBLK_PER_WAVE  8             // 8 blocks per wave
#define WAVES_PER_BLOCK 8            // 256 threads, wave32

// out[row, n*8+j, comp] : comp 0 = real, 1 = imag
// acc columns 0..7 -> out_r(j=0..7), columns 8..15 -> out_i(j=0..7)
__global__ __launch_bounds__(256) void spinor_block_wmma(
    const float* __restrict__ xr, const float* __restrict__ xi,
    const float* __restrict__ wr, const float* __restrict__ wi,
    float* __restrict__ out)
{
  const int lane = threadIdx.x & 31;
  const int wave = threadIdx.x >> 5;
  const int wid  = blockIdx.x * WAVES_PER_BLOCK + wave;

  const int n_groups  = NBLK / NBLK_PER_WAVE;          // 64
  const int row_group = wid / n_groups;                // 0..127
  const int n0        = (wid % n_groups) * NBLK_PER_WAVE;
  const int row_base  = row_group * (TILES_PER_WAVE * 16);

  const int half = lane >> 4;   // 0: lanes 0-15, 1: lanes 16-31
  const int l    = lane & 15;   // row index within M-tile / column index N
  const int cm   = l & 7;       // column within 8-wide block
  const int ci   = l >> 3;      // 0 = real half, 1 = imag half of B

  for (int nn = 0; nn < NBLK_PER_WAVE; ++nn) {
    const int n = n0 + nn;
    const float* __restrict__ wrb = wr + n * (BSZ * BSZ);
    const float* __restrict__ wib = wi + n * (BSZ * BSZ);

    // Build B fragments. Combined B matrices (8x16, split into two 4x16 K-chunks):
    //   B1[k][c] = (c<8) ?  Wr[k][c] : Wi[k][c-8]
    //   B2[k][c] = (c<8) ? -Wi[k][c] : Wr[k][c-8]
    // B 4x16 fragment layout: VGPR v, lane L -> K = v + 2*(L>=16), N = L%16
    v2f b1[2], b2[2];
#pragma unroll
    for (int q = 0; q < 2; ++q) {        // K-chunk: k = q*4 .. q*4+3
#pragma unroll
      for (int v = 0; v < 2; ++v) {
        const int k = q * 4 + v + 2 * half;
        const float wrv = wrb[k * BSZ + cm];
        const float wiv = wib[k * BSZ + cm];
        b1[q][v] = ci ? wiv : wrv;
        b2[q][v] = ci ? wrv : -wiv;
      }
    }

#pragma unroll
    for (int t = 0; t < TILES_PER_WAVE; ++t) {
      // A 16x4 fragment layout: lane L -> M = L%16, VGPR v -> K = v + 2*(L>=16)
      const long row  = (long)(row_base + t * 16 + l);
      const long xoff = row * DIM + n * BSZ;
      const v2f a0r = *(const v2f*)(xr + xoff + 2 * half);      // k 0..3
      const v2f a1r = *(const v2f*)(xr + xoff + 4 + 2 * half);  // k 4..7
      const v2f a0i = *(const v2f*)(xi + xoff + 2 * half);
      const v2f a1i = *(const v2f*)(xi + xoff + 4 + 2 * half);

      v8f acc = {};
      acc = __builtin_amdgcn_wmma_f32_16x16x4_f32(false, a0r, false, b1[0],
                                                  (short)0, acc, false, false);
      acc = __builtin_amdgcn_wmma_f32_16x16x4_f32(false, a1r, false, b1[1],
                                                  (short)0, acc, false, false);
      acc = __builtin_amdgcn_wmma_f32_16x16x4_f32(false, a0i, false, b2[0],
                                                  (short)0, acc, false, false);
      acc = __builtin_amdgcn_wmma_f32_16x16x4_f32(false, a1i, false, b2[1],
                                                  (short)0, acc, false, false);

      // C/D layout: VGPR i, lane L -> M = i + 8*(L>=16), N = L%16
      // out flat index = M_row * (DIM*2) + n*16 + 2*(N%8) + (N/8)
      // Each store instruction writes two complete 64B lines (one per lane half)
      // -> non-temporal store hint: write-once data, keep L2 for the x stream.
      const long obase = (long)(row_base + t * 16 + 8 * half) * (DIM * 2)
                       + (long)n * (BSZ * 2) + 2 * cm + ci;
#pragma unroll
      for (int i = 0; i < 8; ++i) {
        __builtin_nontemporal_store(acc[i], out + obase + (long)i * (DIM * 2));
      }
    }
  }
}

extern "C" void kernel_launch(void* const* d_in, const int* in_sizes, int n_in,
                              void* d_out, int out_size, void* d_ws, size_t ws_size,
                              hipStream_t stream) {
  const float* xr = (const float*)d_in[0];
  const float* xi = (const float*)d_in[1];
  const float* wr = (const float*)d_in[2];
  const float* wi = (const float*)d_in[3];
  float* out = (float*)d_out;

  // waves = (ROWS/64) * (NBLK/8) = 128 * 64 = 8192 ; 8 waves per 256-thread block
  const int total_waves = (ROWS / (TILES_PER_WAVE * 16)) * (NBLK / NBLK_PER_WAVE);
  const int grid = total_waves / WAVES_PER_BLOCK;   // 1024
  spinor_block_wmma<<<grid, WAVES_PER_BLOCK * 32, 0, stream>>>(xr, xi, wr, wi, out);
}